// HHGR_36146444763769
// MI455X (gfx1250) — compile-verified
//
#include <hip/hip_runtime.h>
#include <hip/hip_bf16.h>

typedef float v2f __attribute__((ext_vector_type(2)));
typedef float v8f __attribute__((ext_vector_type(8)));
typedef int   v4i __attribute__((ext_vector_type(4)));

#define GRP  8192
#define MEMB 32
#define DIM  128
#define HATT 16

#ifndef __has_builtin
#define __has_builtin(x) 0
#endif

#if __has_builtin(__builtin_amdgcn_global_load_async_to_lds_b128)
#define HAVE_ASYNC_LDS 1
#else
#define HAVE_ASYNC_LDS 0
#endif

#if __has_builtin(__builtin_amdgcn_s_wait_asynccnt)
#define WAIT_ASYNC(n) __builtin_amdgcn_s_wait_asynccnt(n)
#else
#define WAIT_ASYNC(n) asm volatile("s_wait_asynccnt " #n ::: "memory")
#endif

#if HAVE_ASYNC_LDS
typedef __attribute__((address_space(1))) v4i gv4i_t;  // global int4
typedef __attribute__((address_space(3))) v4i lv4i_t;  // LDS int4

// 16-byte async copy global -> LDS (GLOBAL_LOAD_ASYNC_TO_LDS_B128, ASYNCcnt).
__device__ __forceinline__ void async_cp16(void* lds_dst, const void* gsrc) {
  gv4i_t* g = (gv4i_t*)(uintptr_t)gsrc;
  lv4i_t* l = (lv4i_t*)(uint32_t)(uintptr_t)lds_dst;
  __builtin_amdgcn_global_load_async_to_lds_b128(g, l, 0, 0);
}
#endif

// ---------------------------------------------------------------------------
// Stage 1: per-group member gather + AGREE attention + softmax aggregation.
// One 128-thread block per group. Gather-bandwidth bound; LDS-staged.
// ---------------------------------------------------------------------------
__global__ __launch_bounds__(128)
void hhgr_attention(const int* __restrict__ group_inputs,
                    const int* __restrict__ members,
                    const float* __restrict__ member_mask,
                    const float* __restrict__ user_embedding,
                    const float* __restrict__ user_table,
                    const float* __restrict__ group_table,
                    const float* __restrict__ aw1,
                    const float* __restrict__ ab1,
                    const float* __restrict__ aw2,
                    const float* __restrict__ ab2,
                    float* __restrict__ group_embeds) {
  __shared__ float s_emb[MEMB * DIM];   // 16 KB member embeddings
  __shared__ float s_hid[MEMB * HATT];  //  2 KB hidden activations
  __shared__ float s_log[MEMB];
  __shared__ float s_aw1[DIM * HATT];   //  8 KB attention W1

  const int g = blockIdx.x;
  const int t = threadIdx.x;  // 0..127  (thread == embed dim)

  for (int i = t; i < DIM * HATT; i += 128) s_aw1[i] = aw1[i];

  for (int m = 0; m < MEMB; ++m) {
    const int u = members[g * MEMB + m];
    const size_t off = (size_t)u * DIM + t;
    s_emb[m * DIM + t] = user_table[off] + user_embedding[off];
  }
  __syncthreads();

  for (int task = t; task < MEMB * HATT; task += 128) {
    const int m = task >> 4, h = task & 15;
    float acc = ab1[h];
#pragma unroll 8
    for (int d = 0; d < DIM; ++d)
      acc = fmaf(s_emb[m * DIM + d], s_aw1[d * HATT + h], acc);
    s_hid[m * HATT + h] = fmaxf(acc, 0.0f);
  }
  __syncthreads();

  if (t < MEMB) {
    float l = ab2[0];
#pragma unroll
    for (int h = 0; h < HATT; ++h) l = fmaf(s_hid[t * HATT + h], aw2[h], l);
    s_log[t] = (member_mask[g * MEMB + t] > 0.0f) ? l : -1e9f;
  }
  __syncthreads();

  float mx = -3.4e38f;
#pragma unroll
  for (int m = 0; m < MEMB; ++m) mx = fmaxf(mx, s_log[m]);
  float s = 0.0f, acc = 0.0f;
#pragma unroll
  for (int m = 0; m < MEMB; ++m) {
    const float e = __expf(s_log[m] - mx);
    s += e;
    acc = fmaf(e, s_emb[m * DIM + t], acc);
  }
  const int gi = group_inputs[g];
  group_embeds[(size_t)g * DIM + t] = acc / s + group_table[(size_t)gi * DIM + t];
}

// ---------------------------------------------------------------------------
// Stage 2: C[M x 128] = A[M x K] @ B[K x 128] (optional relu) via fp32 WMMA.
// 256 threads / 8 waves; 32-row M-tile; wave w owns N-block [16w,16w+16) with
// two 16x16 fp32 accumulators.
// Async path: double-buffered KC=32 chunks streamed with
// GLOBAL_LOAD_ASYNC_TO_LDS_B128 overlapping WMMA (ASYNCcnt pipelining).
// Fallback: synchronous KC=64 staging.
// ---------------------------------------------------------------------------
#define LDA 68    // padded LDS row stride for A tile (floats, 272B = 16B-mult)
#define LDB 132   // padded LDS row stride for B tile (floats, 528B = 16B-mult)

__global__ __launch_bounds__(256)
void hhgr_gemm_n128(const float* __restrict__ A, const float* __restrict__ B,
                    float* __restrict__ C, int Mrows, int K, int doRelu) {
  (void)Mrows;
  const int tid  = threadIdx.x;
  const int wave = tid >> 5;
  const int lane = tid & 31;
  const int m0   = blockIdx.x * 32;
  const int nBlk = wave * 16;
  const int lrow = lane & 15;          // A row / B col within fragment
  const int koff = (lane >> 4) << 1;   // lanes 16-31 hold K+2,K+3

  v8f acc0 = {0, 0, 0, 0, 0, 0, 0, 0};
  v8f acc1 = {0, 0, 0, 0, 0, 0, 0, 0};

#if HAVE_ASYNC_LDS
  // ------------------- async double-buffered pipeline --------------------
#define KC2 32
  __shared__ __align__(16) float sA[2][32 * LDA];   // 2 x 8.5 KB
  __shared__ __align__(16) float sB[2][KC2 * LDB];  // 2 x 16.5 KB

  const int arow = tid >> 3, ac4 = (tid & 7) << 2;  // A: 32x32 tile, 1 f4/thr

  // prologue: stream chunk 0 into buffer 0
  async_cp16(&sA[0][arow * LDA + ac4], A + (size_t)(m0 + arow) * K + ac4);
#pragma unroll
  for (int i = 0; i < 4; ++i) {                     // B: 32x128 tile, 4 f4/thr
    const int v = tid + i * 256;
    const int r = v >> 5, c4 = (v & 31) << 2;
    async_cp16(&sB[0][r * LDB + c4], B + (size_t)r * DIM + c4);
  }

  const int nch = K / KC2;
  for (int c = 0; c < nch; ++c) {
    const int bi = c & 1;
    if (c + 1 < nch) {
      const int k0 = (c + 1) * KC2;
      async_cp16(&sA[bi ^ 1][arow * LDA + ac4],
                 A + (size_t)(m0 + arow) * K + k0 + ac4);
#pragma unroll
      for (int i = 0; i < 4; ++i) {
        const int v = tid + i * 256;
        const int r = v >> 5, c4 = (v & 31) << 2;
        async_cp16(&sB[bi ^ 1][r * LDB + c4], B + (size_t)(k0 + r) * DIM + c4);
      }
      WAIT_ASYNC(5);   // allow next chunk's 5 copies in flight; chunk c done
    } else {
      WAIT_ASYNC(0);
    }
    __syncthreads();

    const float* cA = sA[bi];
    const float* cB = sB[bi];
#pragma unroll
    for (int kk = 0; kk < KC2; kk += 4) {
      v2f a0, a1, b;
      a0.x = cA[lrow * LDA + kk + koff];
      a0.y = cA[lrow * LDA + kk + koff + 1];
      a1.x = cA[(lrow + 16) * LDA + kk + koff];
      a1.y = cA[(lrow + 16) * LDA + kk + koff + 1];
      const int n = nBlk + lrow;
      b.x = cB[(kk + koff) * LDB + n];
      b.y = cB[(kk + koff + 1) * LDB + n];
      acc0 = __builtin_amdgcn_wmma_f32_16x16x4_f32(
          false, a0, false, b, (short)0, acc0, false, false);
      acc1 = __builtin_amdgcn_wmma_f32_16x16x4_f32(
          false, a1, false, b, (short)0, acc1, false, false);
    }
    __syncthreads();
  }
#else
  // ------------------- synchronous fallback (KC = 64) --------------------
#define KC 64
  __shared__ float sA[32 * LDA];
  __shared__ float sB[KC * LDB];

  for (int k0 = 0; k0 < K; k0 += KC) {
#pragma unroll
    for (int i = 0; i < 2; ++i) {
      const int v = tid + i * 256;
      const int row = v >> 4, c4 = (v & 15) << 2;
      const float4 d = *(const float4*)(A + (size_t)(m0 + row) * K + k0 + c4);
      float* dst = &sA[row * LDA + c4];
      dst[0] = d.x; dst[1] = d.y; dst[2] = d.z; dst[3] = d.w;
    }
#pragma unroll
    for (int i = 0; i < 8; ++i) {
      const int v = tid + i * 256;
      const int row = v >> 5, c4 = (v & 31) << 2;
      const float4 d = *(const float4*)(B + (size_t)(k0 + row) * DIM + c4);
      float* dst = &sB[row * LDB + c4];
      dst[0] = d.x; dst[1] = d.y; dst[2] = d.z; dst[3] = d.w;
    }
    __syncthreads();
#pragma unroll
    for (int kk = 0; kk < KC; kk += 4) {
      v2f a0, a1, b;
      a0.x = sA[lrow * LDA + kk + koff];
      a0.y = sA[lrow * LDA + kk + koff + 1];
      a1.x = sA[(lrow + 16) * LDA + kk + koff];
      a1.y = sA[(lrow + 16) * LDA + kk + koff + 1];
      const int n = nBlk + lrow;
      b.x = sB[(kk + koff) * LDB + n];
      b.y = sB[(kk + koff + 1) * LDB + n];
      acc0 = __builtin_amdgcn_wmma_f32_16x16x4_f32(
          false, a0, false, b, (short)0, acc0, false, false);
      acc1 = __builtin_amdgcn_wmma_f32_16x16x4_f32(
          false, a1, false, b, (short)0, acc1, false, false);
    }
    __syncthreads();
  }
#endif

  // C/D layout: VGPR r -> M=r (lanes 0-15), M=r+8 (lanes 16-31); N = lane%16
  const int n = nBlk + lrow;
  const int mbase = m0 + ((lane >> 4) << 3);
#pragma unroll
  for (int r = 0; r < 8; ++r) {
    float v0 = acc0[r], v1 = acc1[r];
    if (doRelu) { v0 = fmaxf(v0, 0.0f); v1 = fmaxf(v1, 0.0f); }
    C[(size_t)(mbase + r) * DIM + n] = v0;
    C[(size_t)(mbase + 16 + r) * DIM + n] = v1;
  }
}

// ---------------------------------------------------------------------------
extern "C" void kernel_launch(void* const* d_in, const int* in_sizes, int n_in,
                              void* d_out, int out_size, void* d_ws, size_t ws_size,
                              hipStream_t stream) {
  (void)in_sizes; (void)n_in; (void)out_size; (void)ws_size;

  const int*   group_inputs   = (const int*)d_in[0];
  const int*   members        = (const int*)d_in[1];
  const float* member_mask    = (const float*)d_in[2];
  const float* user_embedding = (const float*)d_in[3];
  const float* H_gl           = (const float*)d_in[4];
  const float* user_table     = (const float*)d_in[5];
  const float* group_table    = (const float*)d_in[6];
  const float* aw1            = (const float*)d_in[7];
  const float* ab1            = (const float*)d_in[8];
  const float* aw2            = (const float*)d_in[9];
  const float* ab2            = (const float*)d_in[10];
  const float* hw1            = (const float*)d_in[11];
  const float* hw2            = (const float*)d_in[12];

  float* ge = (float*)d_ws;                    // group_embeds [G,D]
  float* t1 = ge + (size_t)GRP * DIM;          // group_embeds @ hw1
  float* hb = t1 + (size_t)GRP * DIM;          // relu(H @ t1)
  float* t2 = hb + (size_t)GRP * DIM;          // hb @ hw2
  float* out = (float*)d_out;

  hhgr_attention<<<GRP, 128, 0, stream>>>(
      group_inputs, members, member_mask, user_embedding, user_table,
      group_table, aw1, ab1, aw2, ab2, ge);

  hhgr_gemm_n128<<<GRP / 32, 256, 0, stream>>>(ge, hw1, t1, GRP, DIM, 0);
  hhgr_gemm_n128<<<GRP / 32, 256, 0, stream>>>(H_gl, t1, hb, GRP, GRP, 1);
  hhgr_gemm_n128<<<GRP / 32, 256, 0, stream>>>(hb, hw2, t2, GRP, DIM, 0);
  hhgr_gemm_n128<<<GRP / 32, 256, 0, stream>>>(H_gl, t2, out, GRP, GRP, 0);
}